// GeometricFeatureExtractor_69836168233625
// MI455X (gfx1250) — compile-verified
//
#include <hip/hip_runtime.h>
#include <stdint.h>

#define TPB 256
#define GEPS 1e-10f

// ---- CDNA5 async LDS<->global data movers (gfx1250), per CDNA5 ISA ----
__device__ __forceinline__ void async_load_b128(uint32_t lds_addr, const float* gaddr) {
    // copies 16 bytes from the global address in gaddr to the LDS address in lds_addr  (ASYNCcnt-tracked, streaming hint)
    asm volatile("global_load_async_to_lds_b128 %0, %1, off th:TH_LOAD_NT"
                 :: "v"(lds_addr), "v"(gaddr) : "memory");
}
__device__ __forceinline__ void async_store_b128(float* gaddr, uint32_t lds_addr) {
    // copies 16 bytes from the LDS address in lds_addr to the global address in gaddr  (ASYNCcnt-tracked, streaming hint)
    asm volatile("global_store_async_from_lds_b128 %0, %1, off th:TH_STORE_NT"
                 :: "v"(gaddr), "v"(lds_addr) : "memory");
}
__device__ __forceinline__ void wait_async0() {
    asm volatile("s_wait_asynccnt 0" ::: "memory");
}

__device__ __forceinline__ uint32_t lds_off(const void* p) {
    // generic LDS pointer: low 32 bits are the workgroup-relative LDS byte address
    return (uint32_t)(uintptr_t)p;
}

__global__ __launch_bounds__(TPB) void geom_feat_kernel(
        const float* __restrict__ coords,
        float* __restrict__ angles,   // n*3 floats
        float* __restrict__ frames,   // n*9 floats
        int n) {
    // s_in[j] holds flat float index (base*3 - 4 + j): 194 aligned B128 chunks = 776 floats
    __shared__ __align__(16) float s_in[194 * 4];       // 776
    __shared__ __align__(16) float s_ang[TPB * 3];      // 768: staged angles image
    __shared__ __align__(16) float s_frm[TPB * 9];      // 2304: staged frames image

    const int t    = threadIdx.x;
    const int base = blockIdx.x * TPB;

    // ---------- Phase 1: async-stage coords tile into LDS (194 x B128, 16B-aligned) ----------
    const long long c0 = (long long)base * 3 - 4;       // 16B-aligned flat float start
    if (t < 194) {
        long long gf  = c0 + (long long)t * 4;
        long long lim = (long long)n * 3 - 4;           // last aligned chunk start
        gf = gf < 0 ? 0 : (gf > lim ? lim : gf);        // clamped chunks land in unused slots
        async_load_b128(lds_off(&s_in[t * 4]), coords + gf);
    }
    wait_async0();        // this wave's async loads landed in LDS
    __syncthreads();      // all waves' tiles visible

    // ---------- Phase 2: per-row geometry ----------
    int i  = base + t;
    int iq = i > n - 1 ? n - 1 : i;                     // safety clamp (n % 256 == 0 normally)
    int ic = iq < 1 ? 1 : (iq > n - 2 ? n - 2 : iq);    // stencil center (edge rows copy neighbor)
    // coords row r lives at s_in[(r - base)*3 + 4]; need rows ic-1 .. ic+1
    const float* A = &s_in[(ic - base) * 3 + 1];

    float ax = A[0], ay = A[1], az = A[2];
    float bx = A[3], by = A[4], bz = A[5];
    float cx = A[6], cy = A[7], cz = A[8];

    float d0x = bx - ax, d0y = by - ay, d0z = bz - az;
    float d1x = cx - bx, d1y = cy - by, d1z = cz - bz;

    float n0  = sqrtf(d0x * d0x + d0y * d0y + d0z * d0z);
    float in0 = 1.0f / (n0 + GEPS);
    float v1x = d0x * in0, v1y = d0y * in0, v1z = d0z * in0;

    float n1  = sqrtf(d1x * d1x + d1y * d1y + d1z * d1z);
    float in1 = 1.0f / (n1 + GEPS);
    float v2x = d1x * in1, v2y = d1y * in1, v2z = d1z * in1;

    float dp   = v1x * v2x + v1y * v2y + v1z * v2z;     // unclamped projection
    float ca   = fminf(1.0f, fmaxf(-1.0f, dp));
    float ang  = acosf(ca);

    float wx = v2x - dp * v1x, wy = v2y - dp * v1y, wz = v2z - dp * v1z;
    float n2  = sqrtf(wx * wx + wy * wy + wz * wz);
    float in2 = 1.0f / (n2 + GEPS);
    float e2x = wx * in2, e2y = wy * in2, e2z = wz * in2;

    float e3x = v1y * e2z - v1z * e2y;
    float e3y = v1z * e2x - v1x * e2z;
    float e3z = v1x * e2y - v1y * e2x;

    float av = (i == 0 || i >= n - 1) ? 0.0f : ang;     // angles zero at both ends
    s_ang[t * 3 + 0] = av;
    s_ang[t * 3 + 1] = av;
    s_ang[t * 3 + 2] = av;
    float* F = &s_frm[t * 9];
    F[0] = v1x; F[1] = v1y; F[2] = v1z;
    F[3] = e2x; F[4] = e2y; F[5] = e2z;
    F[6] = e3x; F[7] = e3y; F[8] = e3z;
    __syncthreads();

    // ---------- Phase 3: stream staged images out as coalesced B128 async stores ----------
    const long long a_lim = (long long)n * 3;
    const long long f_lim = (long long)n * 9;

    // angles: 192 x 16B chunks; base*3 floats = base*12 bytes, 16B-aligned (base % 4 == 0)
    if (t < (TPB * 3) / 4) {
        long long gi = (long long)base * 3 + t * 4;
        if (gi + 4 <= a_lim)
            async_store_b128(angles + gi, lds_off(&s_ang[t * 4]));
    }
    // frames: 576 x 16B chunks
#pragma unroll
    for (int k = 0; k < 3; ++k) {
        int c = t + k * TPB;
        if (c < (TPB * 9) / 4) {
            long long gi = (long long)base * 9 + c * 4;
            if (gi + 4 <= f_lim)
                async_store_b128(frames + gi, lds_off(&s_frm[c * 4]));
        }
    }
    // s_endpgm performs an implicit wait-idle: outstanding async stores drain before LDS reuse.
}

extern "C" void kernel_launch(void* const* d_in, const int* in_sizes, int n_in,
                              void* d_out, int out_size, void* d_ws, size_t ws_size,
                              hipStream_t stream) {
    const float* coords = (const float*)d_in[0];
    const int n = in_sizes[0] / 3;                       // 4,000,000 (divisible by 256)
    float* angles = (float*)d_out;                       // first n*3 floats
    float* frames = (float*)d_out + (size_t)n * 3;       // next n*9 floats
    const int grid = (n + TPB - 1) / TPB;
    geom_feat_kernel<<<grid, TPB, 0, stream>>>(coords, angles, frames, n);
}